// FPS_kNN_7103875907739
// MI455X (gfx1250) — compile-verified
//
#include <hip/hip_runtime.h>

#define B_ 8
#define N_ 16384
#define C_ 128
#define S_ 2048
#define K_ 32
#define NCHUNK 512

typedef __attribute__((ext_vector_type(2))) float v2f;
typedef __attribute__((ext_vector_type(8))) float v8f;

// ---- wave32 argmax carrying payload coords (value desc, tie -> lower index) ----
__device__ __forceinline__ void wave_argmax5(float& v, int& i, float& x, float& y, float& z) {
#pragma unroll
  for (int off = 16; off > 0; off >>= 1) {
    float ov = __shfl_xor(v, off, 32);
    int   oi = __shfl_xor(i, off, 32);
    float ox = __shfl_xor(x, off, 32);
    float oy = __shfl_xor(y, off, 32);
    float oz = __shfl_xor(z, off, 32);
    if (ov > v || (ov == v && oi < i)) { v = ov; i = oi; x = ox; y = oy; z = oz; }
  }
}

// ---------------- Kernel 1: furthest point sampling -----------------------------
// One workgroup per batch; each thread owns 16 points in VGPRs (strided by 1024
// for coalesced loads). 2047 sequential argmax rounds; candidate coordinates ride
// along the shuffle reduction so each round needs only 2 block barriers.
__global__ __launch_bounds__(1024) void fps_kernel(const float* __restrict__ xyz,
                                                   int* __restrict__ fps_idx) {
  const int b = blockIdx.x;
  const int t = threadIdx.x;
  const float* xb = xyz + (size_t)b * N_ * 3;

  float px[16], py[16], pz[16], dst[16];
#pragma unroll
  for (int j = 0; j < 16; ++j) {
    int n = j * 1024 + t;
    px[j] = xb[n * 3 + 0];
    py[j] = xb[n * 3 + 1];
    pz[j] = xb[n * 3 + 2];
    dst[j] = 1e10f;
  }

  __shared__ float cs[3];
  __shared__ float wv[32], wx[32], wy[32], wz[32];
  __shared__ int   wi[32];

  if (t == 0) { cs[0] = px[0]; cs[1] = py[0]; cs[2] = pz[0]; fps_idx[b * S_] = 0; }
  __syncthreads();

  const int lane = t & 31, wave = t >> 5;
  for (int s = 1; s < S_; ++s) {
    float cx = cs[0], cy = cs[1], cz = cs[2];
    float lv = -1.0f; int li = 0x7fffffff;
    float lx = 0.f, ly = 0.f, lz = 0.f;
#pragma unroll
    for (int j = 0; j < 16; ++j) {
      float dx = px[j] - cx, dy = py[j] - cy, dz = pz[j] - cz;
      float d = dx * dx + dy * dy + dz * dz;
      d = fminf(dst[j], d);
      dst[j] = d;
      if (d > lv) { lv = d; li = j * 1024 + t; lx = px[j]; ly = py[j]; lz = pz[j]; }
    }
    wave_argmax5(lv, li, lx, ly, lz);
    if (lane == 0) { wv[wave] = lv; wi[wave] = li; wx[wave] = lx; wy[wave] = ly; wz[wave] = lz; }
    __syncthreads();
    if (wave == 0) {
      float v2 = wv[lane]; int i2 = wi[lane];
      float x2 = wx[lane], y2 = wy[lane], z2 = wz[lane];
      wave_argmax5(v2, i2, x2, y2, z2);
      if (lane == 0) {
        fps_idx[b * S_ + s] = i2;
        cs[0] = x2; cs[1] = y2; cs[2] = z2;   // next centroid published directly
      }
    }
    __syncthreads();
  }
}

// ---------------- Kernel 2: kNN via V_WMMA_F32_16X16X4_F32 ----------------------
// Block = 16 waves handles 16 queries. Point chunk is staged to LDS with
// coalesced divergence-free loads; each wave builds B/|p|^2 from LDS (pure
// cndmask selects, EXEC stays all-ones for WMMA). One WMMA yields a full 16x16
// distance tile: D = A(-2q|0) * B(p|0) + C(|q|^2+|p|^2). Selection is
// ballot-filtered top-32 per query-wave + wave bitonic sort (top_k order).
__global__ __launch_bounds__(512) void knn_kernel(const float* __restrict__ xyz,
                                                  const int* __restrict__ fps_idx,
                                                  int* __restrict__ knn_idx) {
  const int tilesPerB = S_ / 16;                 // 128
  const int b = blockIdx.x / tilesPerB;
  const int tile = blockIdx.x % tilesPerB;
  const int s0 = tile * 16;
  const int t = threadIdx.x, lane = t & 31, wave = t >> 5;
  const float* xb = xyz + (size_t)b * N_ * 3;

  __shared__ float qx[16], qy[16], qz[16], q2[16];
  __shared__ float pts[NCHUNK * 3];              // 6 KB staged point coords
  __shared__ float dls[16 * NCHUNK];             // 32 KB distance tile rows

  if (t < 16) {
    int qi = fps_idx[b * S_ + s0 + t];
    float X = xb[qi * 3 + 0], Y = xb[qi * 3 + 1], Z = xb[qi * 3 + 2];
    qx[t] = X; qy[t] = Y; qz[t] = Z; q2[t] = X * X + Y * Y + Z * Z;
  }
  __syncthreads();

  // A operand (16x4 f32): lanes 0-15 hold K=0,1 ; lanes 16-31 hold K=2,3
  v2f a;
  if (lane < 16) { a.x = -2.0f * qx[lane];      a.y = -2.0f * qy[lane]; }
  else           { a.x = -2.0f * qz[lane - 16]; a.y = 0.0f; }
  float q2b[8];
#pragma unroll
  for (int r = 0; r < 8; ++r) q2b[r] = q2[r + 8 * (lane >> 4)];  // C/D: M = r + 8*(lane>=16)

  float bestd = 3.4e38f; int besti = -1;
  float worst = 3.4e38f;

  for (int chunk = 0; chunk < N_ / NCHUNK; ++chunk) {
    const int cbase = chunk * NCHUNK;

    // stage this chunk's coordinates: 512 threads x 3 coalesced b32 loads
#pragma unroll
    for (int e = 0; e < 3; ++e)
      pts[t + e * 512] = xb[cbase * 3 + t + e * 512];
    __syncthreads();

#pragma unroll
    for (int i = 0; i < NCHUNK / 256; ++i) {     // 2 tiles per wave per chunk
      int tl = wave * (NCHUNK / 256) + i;        // local tile 0..31
      int nl = tl * 16 + (lane & 15);            // local point for this lane's N
      float Px = pts[nl * 3 + 0];
      float Py = pts[nl * 3 + 1];
      float Pz = pts[nl * 3 + 2];
      float p2 = Px * Px + Py * Py + Pz * Pz;
      // B operand (4x16): lanes 0-15 -> (px,py), lanes 16-31 -> (pz, 0)
      v2f bm;
      bm.x = (lane < 16) ? Px : Pz;
      bm.y = (lane < 16) ? Py : 0.0f;
      v8f cm;
#pragma unroll
      for (int r = 0; r < 8; ++r) cm[r] = q2b[r] + p2;
      v8f dmat = __builtin_amdgcn_wmma_f32_16x16x4_f32(
          false, a, false, bm, (short)0, cm, false, false);
      int colb = tl * 16 + (lane & 15);
#pragma unroll
      for (int r = 0; r < 8; ++r)
        dls[(r + 8 * (lane >> 4)) * NCHUNK + colb] = dmat[r];
    }
    __syncthreads();

    // selection: wave w scans query w's row for this chunk
    const float* row = &dls[wave * NCHUNK];
    for (int off = 0; off < NCHUNK; off += 32) {
      float d = row[off + lane];
      int gi = cbase + off + lane;
      unsigned rem = 0xffffffffu;
      for (;;) {
        unsigned m = (unsigned)__ballot(d < worst) & rem;
        if (!m) break;
        int src = __ffs(m) - 1;
        rem &= ~(1u << src);
        float cd = __shfl(d, src, 32);
        int   ci = __shfl(gi, src, 32);
        // evict current worst slot
        float mv = bestd; int ml = lane;
#pragma unroll
        for (int o = 16; o > 0; o >>= 1) {
          float omv = __shfl_xor(mv, o, 32);
          int   oml = __shfl_xor(ml, o, 32);
          if (omv > mv || (omv == mv && oml < ml)) { mv = omv; ml = oml; }
        }
        if (lane == ml) { bestd = cd; besti = ci; }
        float w2 = bestd;
#pragma unroll
        for (int o = 16; o > 0; o >>= 1) w2 = fmaxf(w2, __shfl_xor(w2, o, 32));
        worst = w2;
      }
    }
    __syncthreads();
  }

  // bitonic sort ascending by (dist, idx) across the wave
  float d = bestd; int i = besti;
#pragma unroll
  for (int k2 = 2; k2 <= 32; k2 <<= 1) {
#pragma unroll
    for (int j = k2 >> 1; j > 0; j >>= 1) {
      float od = __shfl_xor(d, j, 32);
      int   oi = __shfl_xor(i, j, 32);
      bool dirAsc = ((lane & k2) == 0);
      bool lower  = ((lane & j) == 0);
      bool otherSmaller = (od < d) || (od == d && oi < i);
      bool keepOther = lower ? (dirAsc ? otherSmaller : !otherSmaller)
                             : (dirAsc ? !otherSmaller : otherSmaller);
      if (keepOther) { d = od; i = oi; }
    }
  }
  knn_idx[(b * S_ + s0 + wave) * K_ + lane] = i;
}

// ---------------- Kernel 3/4: gathers (bandwidth-bound) -------------------------
__global__ __launch_bounds__(128) void gather_lc(const float* __restrict__ xyz,
                                                 const float* __restrict__ x,
                                                 const float* __restrict__ rgb,
                                                 const int* __restrict__ fps_idx,
                                                 float* __restrict__ lc_xyz,
                                                 float* __restrict__ lc_x,
                                                 float* __restrict__ lc_rgb) {
  int blk = blockIdx.x;                          // b*S + s
  int b = blk / S_;
  int t = threadIdx.x;
  int idx = fps_idx[blk];
  size_t src = (size_t)(b * N_ + idx);
  lc_x[(size_t)blk * C_ + t] = x[src * C_ + t];
  if (t < 3) {
    lc_xyz[(size_t)blk * 3 + t] = xyz[src * 3 + t];
    lc_rgb[(size_t)blk * 3 + t] = rgb[src * 3 + t];
  }
}

__global__ __launch_bounds__(128) void gather_knn(const float* __restrict__ xyz,
                                                  const float* __restrict__ x,
                                                  const float* __restrict__ rgb,
                                                  const int* __restrict__ knn_idx,
                                                  float* __restrict__ knn_xyz,
                                                  float* __restrict__ knn_x,
                                                  float* __restrict__ knn_rgb) {
  int blk = blockIdx.x;   // b*S*K + s*K + k
  int b = blk / (S_ * K_);
  int t = threadIdx.x;
  int idx = knn_idx[blk];
  size_t src = (size_t)(b * N_ + idx);
  __builtin_prefetch(&x[src * C_ + t], 0, 0);    // global_prefetch_b8
  knn_x[(size_t)blk * C_ + t] = x[src * C_ + t];
  if (t < 3) {
    knn_xyz[(size_t)blk * 3 + t] = xyz[src * 3 + t];
    knn_rgb[(size_t)blk * 3 + t] = rgb[src * 3 + t];
  }
}

// ---------------- launch --------------------------------------------------------
extern "C" void kernel_launch(void* const* d_in, const int* in_sizes, int n_in,
                              void* d_out, int out_size, void* d_ws, size_t ws_size,
                              hipStream_t stream) {
  (void)in_sizes; (void)n_in; (void)out_size; (void)ws_size;
  const float* xyz = (const float*)d_in[0];
  const float* x   = (const float*)d_in[1];
  const float* rgb = (const float*)d_in[2];

  float* out = (float*)d_out;
  float* lc_xyz  = out;                          // [8,2048,3]
  float* lc_x    = out + 49152;                  // [8,2048,128]
  float* lc_rgb  = out + 2146304;                // [8,2048,3]
  float* knn_xyz = out + 2195456;                // [8,2048,32,3]
  float* knn_x   = out + 3768320;                // [8,2048,32,128]
  float* knn_rgb = out + 70877184;               // [8,2048,32,3]

  int* fps_ws = (int*)d_ws;                      // B*S ints
  int* knn_ws = fps_ws + B_ * S_;                // B*S*K ints

  fps_kernel<<<B_, 1024, 0, stream>>>(xyz, fps_ws);
  knn_kernel<<<B_ * (S_ / 16), 512, 0, stream>>>(xyz, fps_ws, knn_ws);
  gather_lc<<<B_ * S_, C_, 0, stream>>>(xyz, x, rgb, fps_ws, lc_xyz, lc_x, lc_rgb);
  gather_knn<<<B_ * S_ * K_, C_, 0, stream>>>(xyz, x, rgb, knn_ws, knn_xyz, knn_x, knn_rgb);
}